// NoDynamicsModel_15247133901110
// MI455X (gfx1250) — compile-verified
//
#include <hip/hip_runtime.h>

typedef __attribute__((ext_vector_type(2))) float v2f;
typedef __attribute__((ext_vector_type(8))) float v8f;

// Grid geometry (shared between kernels and launcher).
constexpr int NBLK = 128;   // blocks -> partials per quantity
constexpr int NTHR = 512;   // threads per block (16 waves of 32)

// ---------------------------------------------------------------------------
// Kernel 1: per-event work.
//   - stage z0 (n_points float2, 80 KB) into LDS (fits: 320 KB / WGP on CDNA5)
//   - grid-stride over events, gather z_i / z_j from LDS, accumulate
//     acc_neg_li = sum(-li) and acc_exp = sum(exp(li)) in registers
//   - LDS tree-reduce the block, emit 2 partial floats per block into d_ws
// ---------------------------------------------------------------------------
extern "C" __global__ void __launch_bounds__(NTHR)
events_partial_kernel(const float* __restrict__ data,
                      const float* __restrict__ beta,
                      const float* __restrict__ z0,
                      float* __restrict__ partials,
                      int n_events, int n_points)
{
    extern __shared__ float lds[];          // n_points float2, reused for reduction
    const int tid = threadIdx.x;

    // ---- stage z0 into LDS as float2 (coalesced 8B loads, ds stores) ----
    const float2* z2 = (const float2*)z0;
    float2* lz = (float2*)lds;
    for (int n = tid; n < n_points; n += NTHR) lz[n] = z2[n];
    __syncthreads();

    const float b = beta[0];
    float acc_neg_li = 0.0f;   // sum of (-li)
    float acc_exp    = 0.0f;   // sum of exp(li)

    const int gstride = NBLK * NTHR;
    for (int e = blockIdx.x * NTHR + tid; e < n_events; e += gstride) {
        // data row: [i, j, t] stored as float; t unused
        const int i = (int)data[3 * e + 0];
        const int j = (int)data[3 * e + 1];
        const float2 zi = lz[i];
        const float2 zj = lz[j];
        const float dx = zi.x - zj.x;
        const float dy = zi.y - zj.y;
        const float li = b - (dx * dx + dy * dy);   // beta - squared distance
        acc_neg_li -= li;
        acc_exp    += expf(li);
    }

    // ---- block tree reduction in LDS (deterministic order) ----
    __syncthreads();                         // done reading lz[]
    lds[tid]        = acc_neg_li;
    lds[NTHR + tid] = acc_exp;
    __syncthreads();
    for (int s = NTHR / 2; s > 0; s >>= 1) {
        if (tid < s) {
            lds[tid]        += lds[tid + s];
            lds[NTHR + tid] += lds[NTHR + tid + s];
        }
        __syncthreads();
    }
    if (tid == 0) {
        partials[blockIdx.x]        = lds[0];      // sum(-li) for this block
        partials[NBLK + blockIdx.x] = lds[NTHR];   // sum(exp(li)) for this block
    }
}

// ---------------------------------------------------------------------------
// Kernel 2: final reduction of 2*NBLK partials with one fp32 WMMA.
// One wave (32 lanes, EXEC all-ones as WMMA requires).
//   A (16x4 f32): lane L -> row M=L%16; a[0],a[1] are (K0,K1) for lanes 0-15
//   and (K2,K3) for lanes 16-31. Feed a[0] = neg_li partial, a[1] = exp partial.
//   B = all ones  =>  D[m][n] = sum_k A[m][k]  (exact: only 2 nonzero terms/row
//   contributed per lane group, fp32 accumulate).
//   Sum of all row-sums = sum over lanes of (neg_li + exp) = answer.
// D layout: VGPR r, lanes 0-15 -> M=r, lanes 16-31 -> M=r+8, so
//   s(lane<16)  = rowsum[0..7],  s(lane>=16) = rowsum[8..15];  shfl_xor(16) joins.
// ---------------------------------------------------------------------------
extern "C" __global__ void __launch_bounds__(32)
wmma_final_reduce_kernel(const float* __restrict__ partials,
                         float* __restrict__ out)
{
    const int lane = threadIdx.x;

    float neg_li = 0.0f, ex = 0.0f;
    for (int k = lane; k < NBLK; k += 32) {
        neg_li += partials[k];
        ex     += partials[NBLK + k];
    }

    v2f a;  a[0] = neg_li; a[1] = ex;
    v2f bb; bb[0] = 1.0f;  bb[1] = 1.0f;
    v8f c = {};
    // 8 args: (neg_a, A, neg_b, B, c_mod, C, reuse_a, reuse_b)
    v8f d = __builtin_amdgcn_wmma_f32_16x16x4_f32(
        false, a, false, bb, (short)0, c, false, false);

    float s = d[0] + d[1] + d[2] + d[3] + d[4] + d[5] + d[6] + d[7];
    s += __shfl_xor(s, 16, 32);             // join the two half-wave row groups

    if (lane == 0) out[0] = s;              // = sum(exp(li)) - sum(li) = -logL
}

// ---------------------------------------------------------------------------
// Launcher. Input order (setup_inputs dict): data, t0, tn, beta, z0.
// ---------------------------------------------------------------------------
extern "C" void kernel_launch(void* const* d_in, const int* in_sizes, int n_in,
                              void* d_out, int out_size, void* d_ws, size_t ws_size,
                              hipStream_t stream) {
    const float* data = (const float*)d_in[0];
    const float* beta = (const float*)d_in[3];
    const float* z0   = (const float*)d_in[4];

    const int n_events = in_sizes[0] / 3;   // 1,000,000
    const int n_points = in_sizes[4] / 2;   // 10,000

    float* partials = (float*)d_ws;         // 2*NBLK floats (1 KB)

    const size_t lds_bytes = (size_t)n_points * sizeof(float2);  // 80 KB
    events_partial_kernel<<<NBLK, NTHR, lds_bytes, stream>>>(
        data, beta, z0, partials, n_events, n_points);

    wmma_final_reduce_kernel<<<1, 32, 0, stream>>>(partials, (float*)d_out);
}